// GCN_16037407883444
// MI455X (gfx1250) — compile-verified
//
#include <hip/hip_runtime.h>

#define NN 50000
#define FDIM 128

typedef __attribute__((ext_vector_type(2))) float v2f;
typedef __attribute__((ext_vector_type(8))) float v8f;

// ---------------- degree / normalization ----------------

__global__ __launch_bounds__(256) void init_deg(float* deg, int n) {
    int i = blockIdx.x * blockDim.x + threadIdx.x;
    if (i < n) deg[i] = 1.0f;                 // self-loop contributes 1 to every node
}

__global__ __launch_bounds__(256) void accum_deg(const int* __restrict__ col,
                                                 float* __restrict__ deg, int e) {
    int i = blockIdx.x * blockDim.x + threadIdx.x;
    if (i < e) atomicAdd(&deg[col[i]], 1.0f);
}

__global__ __launch_bounds__(256) void finalize_dinv(float* d, int n) {
    int i = blockIdx.x * blockDim.x + threadIdx.x;
    if (i < n) d[i] = rsqrtf(d[i]);           // deg >= 1 always, matches jnp.where(deg>0,...)
}

// ---------------- dense transform: H = act(A) @ W  (K = 128) ----------------
// One wave computes a 16-row strip across all FOUT columns using
// V_WMMA_F32_16X16X4_F32. W is staged in LDS once per block.

template <int FOUT, bool RELU_IN>
__global__ __launch_bounds__(256) void gemm_wmma(const float* __restrict__ A,
                                                 const float* __restrict__ W,
                                                 float* __restrict__ H, int nrows) {
    constexpr int K  = FDIM;
    constexpr int CT = FOUT / 16;             // number of 16-wide column tiles
    __shared__ float wlds[K * FOUT];
    for (int i = threadIdx.x; i < K * FOUT; i += 256) wlds[i] = W[i];
    __syncthreads();

    const int wave  = threadIdx.x >> 5;
    const int lane  = threadIdx.x & 31;
    const int strip = blockIdx.x * 8 + wave;
    if (strip >= (nrows >> 4)) return;        // wave-uniform: EXEC stays all-ones

    const int row0 = strip << 4;
    const int m    = lane & 15;               // A: row within tile, B/CD: column
    const int koff = (lane >> 4) << 1;        // lanes 16..31 hold K+2,K+3

    v8f acc[CT];
    const v8f zero = {0.f, 0.f, 0.f, 0.f, 0.f, 0.f, 0.f, 0.f};
#pragma unroll
    for (int t = 0; t < CT; ++t) acc[t] = zero;

    const float* arow = A + (size_t)(row0 + m) * K;

    for (int kk = 0; kk < K; kk += 4) {
        float a0 = arow[kk + koff];
        float a1 = arow[kk + koff + 1];
        if (RELU_IN) { a0 = fmaxf(a0, 0.f); a1 = fmaxf(a1, 0.f); }
        v2f a; a.x = a0; a.y = a1;
#pragma unroll
        for (int t = 0; t < CT; ++t) {
            v2f b;
            b.x = wlds[(kk + koff)     * FOUT + t * 16 + m];
            b.y = wlds[(kk + koff + 1) * FOUT + t * 16 + m];
            acc[t] = __builtin_amdgcn_wmma_f32_16x16x4_f32(
                false, a, false, b, (short)0, acc[t], false, false);
        }
    }

    const int mrow = (lane >> 4) << 3;        // C/D: lanes 16..31 hold rows M+8
#pragma unroll
    for (int t = 0; t < CT; ++t) {
#pragma unroll
        for (int v = 0; v < 8; ++v) {
            H[(size_t)(row0 + mrow + v) * FOUT + t * 16 + m] = acc[t][v];
        }
    }
}

// ---------------- aggregation ----------------
// O[n,f] = b[f] + dinv[n]^2 * H[n,f]   (bias + self-loop; full overwrite, no memset)

template <int F>
__global__ __launch_bounds__(256) void init_out(const float* __restrict__ H,
                                                const float* __restrict__ dinv,
                                                const float* __restrict__ bias,
                                                float* __restrict__ O, int n) {
    long long i = (long long)blockIdx.x * blockDim.x + threadIdx.x;
    if (i < (long long)n * F) {
        int node = (int)(i >> (F == 128 ? 7 : 6));
        int f    = (int)(i & (F - 1));
        float d  = dinv[node];
        O[i] = bias[f] + d * d * H[i];
    }
}

// One wave per edge; each lane handles F/32 contiguous floats -> coalesced
// gather + global_atomic_add_f32 scatter.
template <int F>
__global__ __launch_bounds__(256) void scatter_edges(const float* __restrict__ H,
                                                     const int* __restrict__ row,
                                                     const int* __restrict__ col,
                                                     const float* __restrict__ dinv,
                                                     float* __restrict__ O, int nedges) {
    int gwave = (int)(((long long)blockIdx.x * blockDim.x + threadIdx.x) >> 5);
    int lane  = threadIdx.x & 31;
    if (gwave >= nedges) return;
    int r = row[gwave];
    int c = col[gwave];
    float w = dinv[r] * dinv[c];
    constexpr int VPL = F / 32;
    const float* hr = H + (size_t)r * F + lane * VPL;
    float*       oc = O + (size_t)c * F + lane * VPL;
#pragma unroll
    for (int v = 0; v < VPL; ++v) atomicAdd(&oc[v], w * hr[v]);
}

// ---------------- driver ----------------

extern "C" void kernel_launch(void* const* d_in, const int* in_sizes, int n_in,
                              void* d_out, int out_size, void* d_ws, size_t ws_size,
                              hipStream_t stream) {
    const float* x  = (const float*)d_in[0];
    const int*   ei = (const int*)d_in[1];
    const float* W1 = (const float*)d_in[2];
    const float* b1 = (const float*)d_in[3];
    const float* W2 = (const float*)d_in[4];
    const float* b2 = (const float*)d_in[5];
    const float* W3 = (const float*)d_in[6];
    const float* b3 = (const float*)d_in[7];
    float* out = (float*)d_out;

    const int E = in_sizes[1] / 2;
    const int* row = ei;
    const int* col = ei + E;

    float* ws   = (float*)d_ws;
    float* dinv = ws;                          // NN floats
    float* H    = ws + 50176;                  // NN*128 floats
    float* O    = H + (size_t)NN * FDIM;       // NN*128 floats

    // --- normalization coefficients ---
    init_deg<<<(NN + 255) / 256, 256, 0, stream>>>(dinv, NN);
    accum_deg<<<(E + 255) / 256, 256, 0, stream>>>(col, dinv, E);
    finalize_dinv<<<(NN + 255) / 256, 256, 0, stream>>>(dinv, NN);

    const int ggrid  = ((NN / 16) + 7) / 8;            // 391 blocks, 8 waves each
    const int sgrid  = (int)(((long long)E * 32 + 255) / 256);
    const int ig128  = (int)(((long long)NN * 128 + 255) / 256);
    const int ig64   = (int)(((long long)NN * 64 + 255) / 256);

    // --- layer 1 ---
    gemm_wmma<128, false><<<ggrid, 256, 0, stream>>>(x, W1, H, NN);
    init_out<128><<<ig128, 256, 0, stream>>>(H, dinv, b1, O, NN);
    scatter_edges<128><<<sgrid, 256, 0, stream>>>(H, row, col, dinv, O, E);

    // --- layer 2 (ReLU of layer-1 output fused into A load) ---
    gemm_wmma<128, true><<<ggrid, 256, 0, stream>>>(O, W2, H, NN);
    init_out<128><<<ig128, 256, 0, stream>>>(H, dinv, b2, O, NN);
    scatter_edges<128><<<sgrid, 256, 0, stream>>>(H, row, col, dinv, O, E);

    // --- layer 3, no final activation, writes d_out directly ---
    gemm_wmma<64, true><<<ggrid, 256, 0, stream>>>(O, W3, H, NN);
    init_out<64><<<ig64, 256, 0, stream>>>(H, dinv, b3, out, NN);
    scatter_edges<64><<<sgrid, 256, 0, stream>>>(H, row, col, dinv, out, E);
}